// PoissonSolver_29618094473301
// MI455X (gfx1250) — compile-verified
//
#include <hip/hip_runtime.h>
#include <math.h>

typedef float v2f __attribute__((ext_vector_type(2)));
typedef float v8f __attribute__((ext_vector_type(8)));

#define HID 512
#define TILE 16
#define NCH 5
#define THREADS 256
#define NWAVE 8
#define CTILES (HID / TILE)            // 32 column tiles of 16
#define CT_PER_WAVE (CTILES / NWAVE)   // 4
#define PI_F 3.14159265358979323846f
#define ALPHA_F 0.01f

// LDS activation layout: [ch][k/2][row][2] floats.
// Gives 8B-aligned conflict-free ds_load_b64 for WMMA A fragments.
__device__ __forceinline__ int aoff(int c, int k, int row) {
    return ((c * (HID / 2) + (k >> 1)) * TILE + row) * 2 + (k & 1);
}

__device__ __forceinline__ v8f wmma_f32(v2f a, v2f b, v8f c) {
    // D = A(16x4,f32) * B(4x16,f32) + C(16x16,f32)
    return __builtin_amdgcn_wmma_f32_16x16x4_f32(
        false, a, false, b, (short)0, c, false, false);
}

__global__ __launch_bounds__(THREADS, 1) void pinn_loss_kernel(
    const float* __restrict__ Xi, const float* __restrict__ Xb,
    const float* __restrict__ W0, const float* __restrict__ b0v,
    const float* __restrict__ W1, const float* __restrict__ b1v,
    const float* __restrict__ W2, const float* __restrict__ b2v,
    const float* __restrict__ W3, const float* __restrict__ b3v,
    float* __restrict__ partial, int nI, int nB, int iTiles)
{
    __shared__ float buf[NCH * (HID / 2) * TILE * 2];   // 163840 B = 160 KB

    const int tid  = threadIdx.x;
    const int wave = tid >> 5;
    const int lane = tid & 31;
    const int half = lane >> 4;   // which 16-lane half of the wave
    const int l16  = lane & 15;

    const bool interior = (int)blockIdx.x < iTiles;
    const float* X  = interior ? Xi : Xb;
    const int    n  = interior ? nI : nB;
    const int  tile = interior ? (int)blockIdx.x : ((int)blockIdx.x - iTiles);
    const int  pt0  = tile * TILE;

    // ---------------- layer 0: x(16x2) -> 5 channels (2->512 is VALU-trivial) ----
    for (int idx = tid; idx < TILE * HID; idx += THREADS) {
        const int row = idx & (TILE - 1);
        const int col = idx >> 4;
        int p = pt0 + row; if (p >= n) p = n - 1;   // clamp (tail-safe, excluded later)
        const float px = X[2 * p + 0];
        const float py = X[2 * p + 1];
        const float w0 = W0[col];
        const float w1 = W0[HID + col];
        const float a  = fmaf(px, w0, fmaf(py, w1, b0v[col]));
        const float h  = tanhf(a);
        const float s  = 1.0f - h * h;
        const float m  = -2.0f * h * s;
        buf[aoff(0, col, row)] = h;          // value
        buf[aoff(1, col, row)] = s * w0;     // dh/dx
        buf[aoff(2, col, row)] = s * w1;     // dh/dy
        buf[aoff(3, col, row)] = m * w0 * w0; // d2h/dx2
        buf[aoff(4, col, row)] = m * w1 * w1; // d2h/dy2
    }
    __syncthreads();

    // ---------------- hidden layers 1 and 2 (512x512), in-place 3-phase ---------
    for (int layer = 0; layer < 2; ++layer) {
        const float* __restrict__ W    = layer ? W2  : W1;
        const float* __restrict__ bias = layer ? b2v : b1v;
        // B fragment: lane<16 -> W[k+0..1][n0+l16], lane>=16 -> W[k+2..3][n0+l16]
        const float* WpBase = W + (2 * half) * HID + l16;

        // ---- phase 0: channel 0 (value): pre_a = h_in @ W ; h = tanh(pre_a + b)
        {
            v8f acc[CT_PER_WAVE];
            #pragma unroll
            for (int t = 0; t < CT_PER_WAVE; ++t)
                acc[t] = (v8f){0.f,0.f,0.f,0.f,0.f,0.f,0.f,0.f};
            for (int t = 0; t < CT_PER_WAVE; ++t) {
                const int n0 = (wave * CT_PER_WAVE + t) * TILE;
                const float* Wp = WpBase + n0;
                #pragma unroll 4
                for (int k = 0; k < HID; k += 4) {
                    v2f Bv;
                    Bv.x = Wp[(size_t)(k    ) * HID];
                    Bv.y = Wp[(size_t)(k + 1) * HID];
                    const int kk = k + 2 * half;
                    const v2f A0 = *(const v2f*)&buf[aoff(0, kk, l16)];
                    acc[t] = wmma_f32(A0, Bv, acc[t]);
                }
            }
            __syncthreads();                 // all reads of ch0 done
            #pragma unroll
            for (int t = 0; t < CT_PER_WAVE; ++t) {
                const int N = (wave * CT_PER_WAVE + t) * TILE + l16;
                const float bN = bias[N];
                #pragma unroll
                for (int r = 0; r < 8; ++r) {
                    const int M = r + 8 * half;     // C layout: row = r + 8*half
                    buf[aoff(0, N, M)] = tanhf(acc[t][r] + bN);
                }
            }
            __syncthreads();
        }

        // ---- phase 1: channels 1,2 (raw pre-activation first derivatives) ------
        {
            v8f acc1[CT_PER_WAVE], acc2[CT_PER_WAVE];
            #pragma unroll
            for (int t = 0; t < CT_PER_WAVE; ++t) {
                acc1[t] = (v8f){0.f,0.f,0.f,0.f,0.f,0.f,0.f,0.f};
                acc2[t] = (v8f){0.f,0.f,0.f,0.f,0.f,0.f,0.f,0.f};
            }
            for (int t = 0; t < CT_PER_WAVE; ++t) {
                const int n0 = (wave * CT_PER_WAVE + t) * TILE;
                const float* Wp = WpBase + n0;
                #pragma unroll 4
                for (int k = 0; k < HID; k += 4) {
                    v2f Bv;
                    Bv.x = Wp[(size_t)(k    ) * HID];
                    Bv.y = Wp[(size_t)(k + 1) * HID];
                    const int kk = k + 2 * half;
                    const v2f A1 = *(const v2f*)&buf[aoff(1, kk, l16)];
                    const v2f A2 = *(const v2f*)&buf[aoff(2, kk, l16)];
                    acc1[t] = wmma_f32(A1, Bv, acc1[t]);   // B reused 2x
                    acc2[t] = wmma_f32(A2, Bv, acc2[t]);
                }
            }
            __syncthreads();
            #pragma unroll
            for (int t = 0; t < CT_PER_WAVE; ++t) {
                const int N = (wave * CT_PER_WAVE + t) * TILE + l16;
                #pragma unroll
                for (int r = 0; r < 8; ++r) {
                    const int M = r + 8 * half;
                    buf[aoff(1, N, M)] = acc1[t][r];   // raw da (needed by ddh)
                    buf[aoff(2, N, M)] = acc2[t][r];
                }
            }
            __syncthreads();
        }

        // ---- phase 2: channels 3,4 + finalize 1,2 elementwise ------------------
        {
            v8f acc3[CT_PER_WAVE], acc4[CT_PER_WAVE];
            #pragma unroll
            for (int t = 0; t < CT_PER_WAVE; ++t) {
                acc3[t] = (v8f){0.f,0.f,0.f,0.f,0.f,0.f,0.f,0.f};
                acc4[t] = (v8f){0.f,0.f,0.f,0.f,0.f,0.f,0.f,0.f};
            }
            for (int t = 0; t < CT_PER_WAVE; ++t) {
                const int n0 = (wave * CT_PER_WAVE + t) * TILE;
                const float* Wp = WpBase + n0;
                #pragma unroll 4
                for (int k = 0; k < HID; k += 4) {
                    v2f Bv;
                    Bv.x = Wp[(size_t)(k    ) * HID];
                    Bv.y = Wp[(size_t)(k + 1) * HID];
                    const int kk = k + 2 * half;
                    const v2f A3 = *(const v2f*)&buf[aoff(3, kk, l16)];
                    const v2f A4 = *(const v2f*)&buf[aoff(4, kk, l16)];
                    acc3[t] = wmma_f32(A3, Bv, acc3[t]);
                    acc4[t] = wmma_f32(A4, Bv, acc4[t]);
                }
            }
            __syncthreads();
            #pragma unroll
            for (int t = 0; t < CT_PER_WAVE; ++t) {
                const int N = (wave * CT_PER_WAVE + t) * TILE + l16;
                #pragma unroll
                for (int r = 0; r < 8; ++r) {
                    const int M  = r + 8 * half;
                    const float h   = buf[aoff(0, N, M)];
                    const float s   = 1.0f - h * h;
                    const float m   = -2.0f * h * s;
                    const float pdx = buf[aoff(1, N, M)];   // raw da_x
                    const float pdy = buf[aoff(2, N, M)];   // raw da_y
                    // ddh = s*dda - 2*h*s*da^2 ; dh = s*da (only this thread touches [N,M])
                    buf[aoff(3, N, M)] = fmaf(m, pdx * pdx, s * acc3[t][r]);
                    buf[aoff(4, N, M)] = fmaf(m, pdy * pdy, s * acc4[t][r]);
                    buf[aoff(1, N, M)] = s * pdx;
                    buf[aoff(2, N, M)] = s * pdy;
                }
            }
            __syncthreads();
        }
    }

    // ---------------- final layer (512 -> 1) + per-row loss ---------------------
    {
        const int g = tid >> 4;     // row 0..15
        const int j = tid & 15;     // lane within 16-thread row-group
        float pu = 0.f, pl = 0.f;
        for (int k = j; k < HID; k += 16) {
            const float w = W3[k];
            pu = fmaf(buf[aoff(0, k, g)], w, pu);
            pl = fmaf(buf[aoff(3, k, g)] + buf[aoff(4, k, g)], w, pl);
        }
        #pragma unroll
        for (int mk = 8; mk >= 1; mk >>= 1) {   // half-wave reduce (stays in 16-group)
            pu += __shfl_xor(pu, mk, 32);
            pl += __shfl_xor(pl, mk, 32);
        }
        __syncthreads();            // all LDS reads finished before reuse
        if (j == 0) {
            const int p = pt0 + g;
            float v = 0.f;
            if (p < n) {
                if (interior) {
                    const float px = X[2 * p + 0];
                    const float py = X[2 * p + 1];
                    const float sx = sinf(PI_F * px);
                    const float y2 = py * py;
                    const float f  = -PI_F * PI_F * (1.f + 4.f * y2) * sx * sinf(PI_F * y2)
                                   + 2.f * PI_F * sx * cosf(PI_F * y2);
                    const float r  = pl - f;     // laplacian - forcing
                    v = r * r;
                } else {
                    const float u = pu + b3v[0];
                    v = u * u;
                }
            }
            buf[g] = v;             // reuse start of LDS for 16 row values
        }
        __syncthreads();
        if (tid == 0) {
            float s = 0.f;
            #pragma unroll
            for (int r = 0; r < TILE; ++r) s += buf[r];
            partial[blockIdx.x] = s;    // deterministic per-block partial
        }
    }
}

// Single-thread, fixed-order final reduction: bit-deterministic across replays.
__global__ void pinn_reduce_kernel(const float* __restrict__ partial,
                                   float* __restrict__ out,
                                   int iTiles, int bTiles, int nI, int nB)
{
    if (threadIdx.x == 0 && blockIdx.x == 0) {
        float si = 0.f;
        for (int i = 0; i < iTiles; ++i) si += partial[i];
        float sb = 0.f;
        for (int i = 0; i < bTiles; ++i) sb += partial[iTiles + i];
        const float interior = si / (float)nI;
        const float boundary = sb / (float)nB;
        out[0] = ALPHA_F * interior + boundary;
        out[1] = interior;
        out[2] = boundary;
    }
}

extern "C" void kernel_launch(void* const* d_in, const int* in_sizes, int n_in,
                              void* d_out, int out_size, void* d_ws, size_t ws_size,
                              hipStream_t stream) {
    const float* Xi = (const float*)d_in[0];
    const float* Xb = (const float*)d_in[1];
    const float* W0 = (const float*)d_in[2];
    const float* b0 = (const float*)d_in[3];
    const float* W1 = (const float*)d_in[4];
    const float* b1 = (const float*)d_in[5];
    const float* W2 = (const float*)d_in[6];
    const float* b2 = (const float*)d_in[7];
    const float* W3 = (const float*)d_in[8];
    const float* b3 = (const float*)d_in[9];
    (void)n_in; (void)out_size; (void)ws_size;

    const int nI = in_sizes[0] / 2;
    const int nB = in_sizes[1] / 2;
    const int iTiles = (nI + TILE - 1) / TILE;   // 1250
    const int bTiles = (nB + TILE - 1) / TILE;   // 50
    float* partial = (float*)d_ws;               // (iTiles+bTiles) floats

    pinn_loss_kernel<<<dim3(iTiles + bTiles), dim3(THREADS), 0, stream>>>(
        Xi, Xb, W0, b0, W1, b1, W2, b2, W3, b3, partial, nI, nB, iTiles);
    pinn_reduce_kernel<<<dim3(1), dim3(1), 0, stream>>>(
        partial, (float*)d_out, iTiles, bTiles, nI, nB);
}